// HardAttentionLayer_13993003450659
// MI455X (gfx1250) — compile-verified
//
#include <hip/hip_runtime.h>
#include <math.h>

typedef __attribute__((ext_vector_type(2))) float v2f;
typedef __attribute__((ext_vector_type(8))) float v8f;

constexpr int BB  = 512;   // batch
constexpr int SEQ = 100;   // sequence
constexpr int HID = 1024;  // hidden
constexpr int NH  = 8;     // heads
constexpr int AH  = 128;   // head dim (A)
constexpr float LOGIT_SCALE = 1.0f / (32.0f * 100.0f); // 1/(sqrt(1024)*S)

// ---------------------------------------------------------------------------
// Kernel A: positional encoding table pe[S,H] and its column sums pesum[H]
// ---------------------------------------------------------------------------
__global__ __launch_bounds__(256) void pe_kernel(float* __restrict__ pe,
                                                 float* __restrict__ pesum) {
  int h = blockIdx.x * 256 + threadIdx.x;
  if (h >= HID) return;
  float div = __expf((float)((h >> 1) * 2) * (-logf(10000.0f) / (float)HID));
  float acc = 0.0f;
  for (int s = 0; s < SEQ; ++s) {
    float ang = (float)s * div;
    float v = (h & 1) ? __cosf(ang) : __sinf(ang);
    pe[s * HID + h] = v;
    acc += v;
  }
  pesum[h] = acc;
}

// ---------------------------------------------------------------------------
// Kernel B: xsum[b,h] = sum_s x[b,s,h] + pesum[h]   (== sum_s pos_emb[b,s,h])
// ---------------------------------------------------------------------------
__global__ __launch_bounds__(256) void xsum_kernel(const float* __restrict__ x,
                                                   const float* __restrict__ pesum,
                                                   float* __restrict__ xsum) {
  int idx = blockIdx.x * 256 + threadIdx.x;   // B*H threads
  int h = idx & (HID - 1);
  float acc = pesum[h];
  const float* xp = x + (size_t)(idx >> 10) * SEQ * HID + h;
  for (int s = 0; s < SEQ; ++s) acc += xp[s * HID];
  xsum[idx] = acc;
}

// ---------------------------------------------------------------------------
// Kernel C: Ksum[b,j] = sum_h xsum[b,h] * Wk[j,h]   (M=512, N=1024, K=1024)
// WMMA f32 16x16x4; each wave computes a 16(M) x 64(N) tile.
// A lane layout: row = m0+(lane&15), k = k0+2*(lane>>4) -> contiguous float2
// B lane layout: col = n0+(lane&15), k = k0+2*(lane>>4) -> contiguous float2
// (both xsum and Wk are K-major per row, so both fragments are b64 loads)
// ---------------------------------------------------------------------------
__global__ __launch_bounds__(128) void ksum_gemm(const float* __restrict__ xsum,
                                                 const float* __restrict__ Wk,
                                                 float* __restrict__ Ksum) {
  const int wid    = (blockIdx.x * 128 + threadIdx.x) >> 5;  // 512 waves
  const int lane   = threadIdx.x & 31;
  const int lane16 = lane & 15;
  const int khalf  = lane >> 4;
  const int m0 = (wid & 31) * 16;   // 32 M tiles
  const int n0 = (wid >> 5) * 64;   // 16 N super-tiles of 64

  v8f acc0 = {}, acc1 = {}, acc2 = {}, acc3 = {};
  const float* aptr = xsum + (size_t)(m0 + lane16) * HID + 2 * khalf;
  const float* bptr = Wk   + (size_t)(n0 + lane16) * HID + 2 * khalf;

  for (int k0 = 0; k0 < HID; k0 += 4) {
    __builtin_prefetch(aptr + k0 + 256, 0, 1);
    v2f a  = *(const v2f*)(aptr + k0);
    v2f b0 = *(const v2f*)(bptr + k0);
    v2f b1 = *(const v2f*)(bptr + 16 * HID + k0);
    v2f b2 = *(const v2f*)(bptr + 32 * HID + k0);
    v2f b3 = *(const v2f*)(bptr + 48 * HID + k0);
    acc0 = __builtin_amdgcn_wmma_f32_16x16x4_f32(false, a, false, b0, (short)0, acc0, false, false);
    acc1 = __builtin_amdgcn_wmma_f32_16x16x4_f32(false, a, false, b1, (short)0, acc1, false, false);
    acc2 = __builtin_amdgcn_wmma_f32_16x16x4_f32(false, a, false, b2, (short)0, acc2, false, false);
    acc3 = __builtin_amdgcn_wmma_f32_16x16x4_f32(false, a, false, b3, (short)0, acc3, false, false);
  }

#pragma unroll
  for (int i = 0; i < 8; ++i) {
    int row = m0 + i + 8 * khalf;       // C/D layout: VGPR i -> M=i (+8 for hi half)
    float* cp = Ksum + (size_t)row * HID + n0 + lane16;
    cp[0]  = acc0[i];
    cp[16] = acc1[i];
    cp[32] = acc2[i];
    cp[48] = acc3[i];
  }
}

// ---------------------------------------------------------------------------
// Kernel D: w[b,n,h] = sum_a Ksum[b, n*A+a] * Wq[n*A+a, h]
// Per head: M=512(b), K=128(a), N=1024(h). Wq is N-major per K row, so the
// B fragment is two strided dword loads per lane.
// ---------------------------------------------------------------------------
__global__ __launch_bounds__(128) void w_gemm(const float* __restrict__ Ksum,
                                              const float* __restrict__ Wq,
                                              float* __restrict__ wbuf) {
  const int wid    = (blockIdx.x * 128 + threadIdx.x) >> 5;  // 4096 waves
  const int lane   = threadIdx.x & 31;
  const int lane16 = lane & 15;
  const int khalf  = lane >> 4;
  const int head = wid >> 9;          // 512 waves per head
  const int t    = wid & 511;
  const int m0 = (t & 31) * 16;
  const int h0 = (t >> 5) * 64;

  v8f acc0 = {}, acc1 = {}, acc2 = {}, acc3 = {};
  const float* aptr  = Ksum + (size_t)(m0 + lane16) * HID + head * AH + 2 * khalf;
  const float* bbase = Wq + (size_t)(head * AH + 2 * khalf) * HID + h0 + lane16;

  for (int k0 = 0; k0 < AH; k0 += 4) {
    v2f a = *(const v2f*)(aptr + k0);
    const float* bp = bbase + (size_t)k0 * HID;
    v2f b0 = { bp[0],  bp[HID + 0]  };
    v2f b1 = { bp[16], bp[HID + 16] };
    v2f b2 = { bp[32], bp[HID + 32] };
    v2f b3 = { bp[48], bp[HID + 48] };
    acc0 = __builtin_amdgcn_wmma_f32_16x16x4_f32(false, a, false, b0, (short)0, acc0, false, false);
    acc1 = __builtin_amdgcn_wmma_f32_16x16x4_f32(false, a, false, b1, (short)0, acc1, false, false);
    acc2 = __builtin_amdgcn_wmma_f32_16x16x4_f32(false, a, false, b2, (short)0, acc2, false, false);
    acc3 = __builtin_amdgcn_wmma_f32_16x16x4_f32(false, a, false, b3, (short)0, acc3, false, false);
  }

#pragma unroll
  for (int i = 0; i < 8; ++i) {
    int row = m0 + i + 8 * khalf;
    float* cp = wbuf + ((size_t)row * NH + head) * HID + h0 + lane16;
    cp[0]  = acc0[i];
    cp[16] = acc1[i];
    cp[32] = acc2[i];
    cp[48] = acc3[i];
  }
}

// ---------------------------------------------------------------------------
// Kernel E: logits[b,n,s] = (x[b,s,:]+pe[s,:]) . w[b,n,:] * scale ; then
// y = logits + gumbel, softmax+argmax over s, straight-through alphas.
// One block per batch b; w[b] staged in LDS (32 KB); wave i handles s=i,i+8,..
// then wave n does head-n softmax (wave32 shuffles).
// ---------------------------------------------------------------------------
__global__ __launch_bounds__(256) void logits_alphas(const float* __restrict__ x,
                                                     const float* __restrict__ pe,
                                                     const float* __restrict__ wbuf,
                                                     const float* __restrict__ gumbel,
                                                     float* __restrict__ alphas) {
  __shared__ float wlds[NH * HID];   // 32 KB
  __shared__ float lg[NH][SEQ];
  const int b = blockIdx.x;

  for (int i = threadIdx.x; i < NH * HID; i += 256)
    wlds[i] = wbuf[(size_t)b * NH * HID + i];
  __syncthreads();

  const int wave = threadIdx.x >> 5;
  const int lane = threadIdx.x & 31;

  for (int s = wave; s < SEQ; s += NH) {
    const float* xr = x + ((size_t)b * SEQ + s) * HID;
    const float* pr = pe + (size_t)s * HID;
    float acc[NH];
#pragma unroll
    for (int n = 0; n < NH; ++n) acc[n] = 0.0f;
    for (int j = lane; j < HID; j += 32) {
      float v = xr[j] + pr[j];
#pragma unroll
      for (int n = 0; n < NH; ++n) acc[n] += v * wlds[n * HID + j];
    }
#pragma unroll
    for (int n = 0; n < NH; ++n) {
      float r = acc[n];
      for (int off = 16; off > 0; off >>= 1) r += __shfl_down(r, off, 32);
      if (lane == 0) lg[n][s] = r;
    }
  }
  __syncthreads();

  // softmax + straight-through per head (wave n -> head n)
  const int n = wave;
  float y[4];
  float mymax = -__builtin_inff();
  int   myarg = 0;
#pragma unroll
  for (int q = 0; q < 4; ++q) {
    int s = lane + 32 * q;
    if (s < SEQ) {
      float v = lg[n][s] * LOGIT_SCALE + gumbel[((size_t)b * NH + n) * SEQ + s];
      y[q] = v;
      if (v > mymax) { mymax = v; myarg = s; }
    } else {
      y[q] = -__builtin_inff();
    }
  }
  for (int off = 16; off > 0; off >>= 1) {
    float om = __shfl_down(mymax, off, 32);
    int   oa = __shfl_down(myarg, off, 32);
    if (om > mymax || (om == mymax && oa < myarg)) { mymax = om; myarg = oa; }
  }
  mymax = __shfl(mymax, 0, 32);
  myarg = __shfl(myarg, 0, 32);

  float e[4];
  float esum = 0.0f;
#pragma unroll
  for (int q = 0; q < 4; ++q) {
    int s = lane + 32 * q;
    e[q] = (s < SEQ) ? __expf(y[q] - mymax) : 0.0f;
    esum += e[q];
  }
  for (int off = 16; off > 0; off >>= 1) esum += __shfl_down(esum, off, 32);
  esum = __shfl(esum, 0, 32);
  float inv = 1.0f / esum;
#pragma unroll
  for (int q = 0; q < 4; ++q) {
    int s = lane + 32 * q;
    if (s < SEQ) {
      float soft = e[q] * inv;
      float hard = (s == myarg) ? 1.0f : 0.0f;
      alphas[((size_t)b * NH + n) * SEQ + s] = (hard - soft) + soft;
    }
  }
}

// ---------------------------------------------------------------------------
// Kernel F: out[b,n,h] = sum_s alphas[b,n,s] * x[b,s,h]
// One block per b. alphas[b] staged into LDS with CDNA5 async global->LDS
// loads (ASYNCcnt), then each thread owns a float4 column slice for all 8
// heads so x[b] streams from HBM exactly once.
// ---------------------------------------------------------------------------
__global__ __launch_bounds__(256) void gather_out(const float* __restrict__ x,
                                                  const float* __restrict__ alphas,
                                                  float* __restrict__ out) {
  __shared__ float als[NH * SEQ];    // 800 floats
  const int b = blockIdx.x;

  for (int i = threadIdx.x; i < NH * SEQ; i += 256) {
    unsigned lds_off = (unsigned)(size_t)(&als[i]);
    const float* g = alphas + (size_t)b * NH * SEQ + i;
    asm volatile("global_load_async_to_lds_b32 %0, %1, off"
                 :: "v"(lds_off), "v"(g) : "memory");
  }
  asm volatile("s_wait_asynccnt 0x0" ::: "memory");
  __syncthreads();

  const int h4 = threadIdx.x * 4;    // 256 threads * 4 = 1024 = H
  float4 acc[NH];
#pragma unroll
  for (int n = 0; n < NH; ++n) acc[n] = make_float4(0.f, 0.f, 0.f, 0.f);

  const float* xb = x + (size_t)b * SEQ * HID + h4;
  for (int s = 0; s < SEQ; ++s) {
    const float4 xv = *(const float4*)(xb + (size_t)s * HID);
#pragma unroll
    for (int n = 0; n < NH; ++n) {
      float a = als[n * SEQ + s];
      acc[n].x += a * xv.x;
      acc[n].y += a * xv.y;
      acc[n].z += a * xv.z;
      acc[n].w += a * xv.w;
    }
  }
#pragma unroll
  for (int n = 0; n < NH; ++n)
    *(float4*)(out + ((size_t)b * NH + n) * HID + h4) = acc[n];
}

// ---------------------------------------------------------------------------
extern "C" void kernel_launch(void* const* d_in, const int* in_sizes, int n_in,
                              void* d_out, int out_size, void* d_ws, size_t ws_size,
                              hipStream_t stream) {
  const float* x      = (const float*)d_in[0];
  const float* Wq     = (const float*)d_in[1];
  const float* Wk     = (const float*)d_in[2];
  const float* gumbel = (const float*)d_in[3];
  float* out = (float*)d_out;

  // workspace layout (floats): pe | pesum | xsum | Ksum | w | alphas  (~23 MB)
  float* pe     = (float*)d_ws;
  float* pesum  = pe + (size_t)SEQ * HID;
  float* xsum   = pesum + HID;
  float* Ksum   = xsum + (size_t)BB * HID;
  float* wbuf   = Ksum + (size_t)BB * HID;
  float* alphas = wbuf + (size_t)BB * NH * HID;

  pe_kernel    <<<(HID + 255) / 256, 256, 0, stream>>>(pe, pesum);
  xsum_kernel  <<<(BB * HID) / 256,  256, 0, stream>>>(x, pesum, xsum);
  ksum_gemm    <<<128,  128, 0, stream>>>(xsum, Wk, Ksum);
  w_gemm       <<<1024, 128, 0, stream>>>(Ksum, Wq, wbuf);
  logits_alphas<<<BB,   256, 0, stream>>>(x, pe, wbuf, gumbel, alphas);
  gather_out   <<<BB,   256, 0, stream>>>(x, alphas, out);
}